// Social_Aggregator_53867479826638
// MI455X (gfx1250) — compile-verified
//
#include <hip/hip_runtime.h>
#include <hip/hip_bf16.h>

typedef float v2f __attribute__((ext_vector_type(2)));
typedef float v8f __attribute__((ext_vector_type(8)));

#define EMBED 128

// ---------------------------------------------------------------------------
// Kernel 1: per-edge attention scores via fp32 WMMA.
// One wave (32 lanes) computes a 16-edge tile:
//   x[16,256] -> h1[16,128] -> h2[16,128] -> s[16]
// using V_WMMA_F32_16X16X4_F32 (D = A(16x4) * B(4x16) + C).
// Fragment layouts (ISA 7.12.2, rule: row = vgpr + (rows/2)*half):
//   A: lane = half*16 + m ; vgpr j holds x[m][k0 + 2*half + j]
//   B: lane = half*16 + n ; vgpr j holds W^T[k0 + 2*half + j][n]
//   C: lane = half*16 + n ; vgpr r holds h[r + 8*half][n]
// ---------------------------------------------------------------------------
__global__ __launch_bounds__(64)
void social_attn_scores(const int* __restrict__ nodes,
                        const int* __restrict__ neigh_idx,
                        const int* __restrict__ seg_ids,
                        const float* __restrict__ u2e,
                        const float* __restrict__ att1_w,   // [128,256] row-major
                        const float* __restrict__ att1_b,   // [128]
                        const float* __restrict__ att2_w,   // [128,128]
                        const float* __restrict__ att2_b,   // [128]
                        const float* __restrict__ att3_w,   // [128]
                        const float* __restrict__ att3_b,   // [1]
                        float* __restrict__ s_out,
                        int E)
{
    // stride 260 floats: 260*4 = 1040 bytes (16B aligned rows),
    // 260 % 64 = 4 dword-banks per row -> 16 rows hit 16 distinct banks.
    __shared__ float xt[2][16][260];
    __shared__ float sred[2][32][8];

    const int lane = threadIdx.x & 31;
    const int wv   = threadIdx.x >> 5;
    const int m16  = lane & 15;
    const int half = lane >> 4;

    const int tile = blockIdx.x * 2 + wv;
    const int eb   = tile * 16;

    // ---- stage x tile: lanes 0-15 load neighbor half, 16-31 center half ----
    {
        int e = eb + m16;
        if (e >= E) e = E - 1;
        const float* src = (half == 0)
            ? u2e + (size_t)neigh_idx[e] * EMBED
            : u2e + (size_t)nodes[seg_ids[e]] * EMBED;
        float4* dst = (float4*)&xt[wv][m16][half * EMBED];
        #pragma unroll
        for (int i = 0; i < EMBED / 4; ++i)
            dst[i] = ((const float4*)src)[i];
    }
    __syncthreads();

    // ---- GEMM1: h1[16,128] = relu(x @ W1^T + b1), K = 256 ----
    v8f acc[8];
    #pragma unroll
    for (int nt = 0; nt < 8; ++nt) {
        float b = att1_b[nt * 16 + m16];            // C depends only on N
        #pragma unroll
        for (int r = 0; r < 8; ++r) acc[nt][r] = b;
    }
    for (int k0 = 0; k0 < 2 * EMBED; k0 += 4) {
        const int kb = k0 + half * 2;
        v2f a;
        a.x = xt[wv][m16][kb];
        a.y = xt[wv][m16][kb + 1];
        #pragma unroll
        for (int nt = 0; nt < 8; ++nt) {
            const float* bp = att1_w + (size_t)(nt * 16 + m16) * (2 * EMBED) + kb;
            v2f b; b.x = bp[0]; b.y = bp[1];        // B[k][n] = att1_w[n][k]
            acc[nt] = __builtin_amdgcn_wmma_f32_16x16x4_f32(
                false, a, false, b, (short)0, acc[nt], false, false);
        }
    }
    __syncthreads();

    // relu + reshape C-layout -> LDS rows so GEMM2 can read A-fragments
    #pragma unroll
    for (int nt = 0; nt < 8; ++nt) {
        #pragma unroll
        for (int r = 0; r < 8; ++r) {
            float h = acc[nt][r];
            h = h > 0.0f ? h : 0.0f;
            xt[wv][r + 8 * half][nt * 16 + m16] = h;
        }
    }
    __syncthreads();

    // ---- GEMM2: h2[16,128] = relu(h1 @ W2^T + b2), K = 128 ----
    v8f acc2[8];
    #pragma unroll
    for (int nt = 0; nt < 8; ++nt) {
        float b = att2_b[nt * 16 + m16];
        #pragma unroll
        for (int r = 0; r < 8; ++r) acc2[nt][r] = b;
    }
    for (int k0 = 0; k0 < EMBED; k0 += 4) {
        const int kb = k0 + half * 2;
        v2f a;
        a.x = xt[wv][m16][kb];
        a.y = xt[wv][m16][kb + 1];
        #pragma unroll
        for (int nt = 0; nt < 8; ++nt) {
            const float* bp = att2_w + (size_t)(nt * 16 + m16) * EMBED + kb;
            v2f b; b.x = bp[0]; b.y = bp[1];
            acc2[nt] = __builtin_amdgcn_wmma_f32_16x16x4_f32(
                false, a, false, b, (short)0, acc2[nt], false, false);
        }
    }

    // ---- s[m] = sum_n relu(h2[m][n]) * att3_w[n] + b3 ----
    float ps[8];
    #pragma unroll
    for (int r = 0; r < 8; ++r) ps[r] = 0.0f;
    #pragma unroll
    for (int nt = 0; nt < 8; ++nt) {
        float w3 = att3_w[nt * 16 + m16];
        #pragma unroll
        for (int r = 0; r < 8; ++r) {
            float h = acc2[nt][r];
            h = h > 0.0f ? h : 0.0f;
            ps[r] += h * w3;                         // partial over N = this lane's column
        }
    }
    #pragma unroll
    for (int r = 0; r < 8; ++r) sred[wv][lane][r] = ps[r];
    __syncthreads();

    if (lane < 16) {
        const int m  = lane;
        const int l0 = (m < 8) ? 0 : 16;             // rows 0-7 live in lanes 0-15
        float s = att3_b[0];
        #pragma unroll
        for (int j = 0; j < 16; ++j) s += sred[wv][l0 + j][m & 7];
        const int e = eb + m;
        if (e < E) s_out[e] = s;
    }
}

// ---------------------------------------------------------------------------
// Kernel 2: segment softmax + weighted aggregation, one block per node.
// seg_ids is sorted, so each thread binary-searches its node's edge range.
// Thread t owns output dimension t (coalesced u2e gathers across the block).
// ---------------------------------------------------------------------------
__global__ __launch_bounds__(128)
void social_agg(const int* __restrict__ nodes,
                const int* __restrict__ neigh_idx,
                const int* __restrict__ seg_ids,
                const float* __restrict__ u2e,
                const float* __restrict__ s,
                float* __restrict__ out,
                int E)
{
    const int node = blockIdx.x;
    const int d    = threadIdx.x;

    // lower_bound(seg_ids, node)
    int lo = 0, hi = E;
    while (lo < hi) {
        int mid = (lo + hi) >> 1;
        if (seg_ids[mid] < node) lo = mid + 1; else hi = mid;
    }
    const int start = lo;
    // upper_bound(seg_ids, node)
    hi = E;
    while (lo < hi) {
        int mid = (lo + hi) >> 1;
        if (seg_ids[mid] <= node) lo = mid + 1; else hi = mid;
    }
    const int end = lo;

    if (start == end) {                              // zero-degree fallback
        out[(size_t)node * EMBED + d] = u2e[(size_t)nodes[node] * EMBED + d];
        return;
    }

    float m = -3.402823466e38f;
    for (int e = start; e < end; ++e)
        m = fmaxf(m, s[e]);                          // broadcast loads (L2 hit)

    float denom = 0.0f;
    float acc   = 0.0f;
    for (int e = start; e < end; ++e) {
        float w = expf(s[e] - m);
        denom += w;
        acc   += w * u2e[(size_t)neigh_idx[e] * EMBED + d];
    }
    out[(size_t)node * EMBED + d] = acc / denom;
}

extern "C" void kernel_launch(void* const* d_in, const int* in_sizes, int n_in,
                              void* d_out, int out_size, void* d_ws, size_t ws_size,
                              hipStream_t stream) {
    const int*   nodes     = (const int*)  d_in[0];
    const int*   neigh_idx = (const int*)  d_in[1];
    const int*   seg_ids   = (const int*)  d_in[2];
    const float* u2e       = (const float*)d_in[3];
    const float* att1_w    = (const float*)d_in[4];
    const float* att1_b    = (const float*)d_in[5];
    const float* att2_w    = (const float*)d_in[6];
    const float* att2_b    = (const float*)d_in[7];
    const float* att3_w    = (const float*)d_in[8];
    const float* att3_b    = (const float*)d_in[9];

    float* out = (float*)d_out;
    float* s   = (float*)d_ws;       // E floats of scratch for edge scores

    const int E = in_sizes[1];       // neigh_idx count
    const int N = in_sizes[0];       // node count

    const int tiles = (E + 15) / 16;
    const int grid1 = (tiles + 1) / 2;    // 2 waves (tiles) per block

    social_attn_scores<<<grid1, 64, 0, stream>>>(
        nodes, neigh_idx, seg_ids, u2e,
        att1_w, att1_b, att2_w, att2_b, att3_w, att3_b, s, E);

    social_agg<<<N, 128, 0, stream>>>(nodes, neigh_idx, seg_ids, u2e, s, out, E);
}